// PointNetFeaturePropagation_28982439313473
// MI455X (gfx1250) — compile-verified
//
#include <hip/hip_runtime.h>
#include <hip/hip_bf16.h>
#include <math.h>

typedef __attribute__((ext_vector_type(2))) float v2f;
typedef __attribute__((ext_vector_type(8))) float v8f;

#define BATCH 16
#define NPTS  4096
#define SPTS  1024
#define DATA_DIM 253
#define IN_CH 256           // 253 + 3
#define ROWS (BATCH * NPTS) // 65536
#define C0 256
#define C1 128
#define BN_EPS 1e-5f

// ---- workspace layout (in floats) ----
#define F_INTERP 0u                       // 65536*3
#define F_STATS  (ROWS * 3u)              // 196608
#define F_SUM0   (F_STATS + 0u)           // 256
#define F_SQ0    (F_STATS + 256u)         // 256
#define F_SCALE0 (F_STATS + 512u)         // 256
#define F_SHIFT0 (F_STATS + 768u)         // 256
#define F_SUM1   (F_STATS + 1024u)        // 128
#define F_SQ1    (F_STATS + 1152u)        // 128
#define F_SCALE1 (F_STATS + 1280u)        // 128
#define F_SHIFT1 (F_STATS + 1408u)        // 128
#define F_Y0     (F_STATS + 1536u)        // 65536*256

// ------------------------------------------------------------------
// 0) zero the stat accumulators
// ------------------------------------------------------------------
__global__ void zero_stats_kernel(float* ws) {
    for (unsigned k = threadIdx.x; k < 1536u; k += blockDim.x)
        ws[F_STATS + k] = 0.0f;
}

// ------------------------------------------------------------------
// 1) 3-NN inverse-distance interpolation of sample_coords
// ------------------------------------------------------------------
__global__ __launch_bounds__(128) void interp_kernel(
    const float* __restrict__ coords,         // [B,N,3]
    const float* __restrict__ sample_coords,  // [B,S,3]
    float* __restrict__ interp)               // [B*N,3]
{
    __shared__ float scx[SPTS], scy[SPTS], scz[SPTS];
    const int b = blockIdx.x;
    const int n = blockIdx.y * 128 + threadIdx.x;
    const float* sc = sample_coords + (size_t)b * SPTS * 3;
    for (int i = threadIdx.x; i < SPTS; i += 128) {
        scx[i] = sc[i * 3 + 0];
        scy[i] = sc[i * 3 + 1];
        scz[i] = sc[i * 3 + 2];
    }
    __syncthreads();

    const size_t row = (size_t)b * NPTS + n;
    const float cx = coords[row * 3 + 0];
    const float cy = coords[row * 3 + 1];
    const float cz = coords[row * 3 + 2];
    const float cn = cx * cx + cy * cy + cz * cz;

    float d0 = INFINITY, d1 = INFINITY, d2 = INFINITY;
    int i0 = 0, i1 = 0, i2 = 0;
    for (int s = 0; s < SPTS; ++s) {
        const float sx = scx[s], sy = scy[s], sz = scz[s];
        const float sn = sx * sx + sy * sy + sz * sz;
        const float d = cn + sn - 2.0f * (cx * sx + cy * sy + cz * sz);
        if (d < d0)      { d2 = d1; i2 = i1; d1 = d0; i1 = i0; d0 = d; i0 = s; }
        else if (d < d1) { d2 = d1; i2 = i1; d1 = d;  i1 = s; }
        else if (d < d2) { d2 = d;  i2 = s; }
    }
    const float r0 = 1.0f / (d0 + 1e-8f);
    const float r1 = 1.0f / (d1 + 1e-8f);
    const float r2 = 1.0f / (d2 + 1e-8f);
    const float rs = 1.0f / (r0 + r1 + r2);
    const float w0 = r0 * rs, w1 = r1 * rs, w2 = r2 * rs;
    interp[row * 3 + 0] = w0 * scx[i0] + w1 * scx[i1] + w2 * scx[i2];
    interp[row * 3 + 1] = w0 * scy[i0] + w1 * scy[i1] + w2 * scy[i2];
    interp[row * 3 + 2] = w0 * scz[i0] + w1 * scz[i1] + w2 * scz[i2];
}

// ------------------------------------------------------------------
// WMMA f32 16x16x4 GEMM core.
// Block = 256 threads = 8 waves; block tile M=128 x N=128.
// Wave (wm 0..3, wn 0..1): rows [wm*32, +32) (two 16-row A frags),
// cols [wn*64, +64) (four 16-col B frags)  -> 8 accumulator tiles.
// LDS: As[row][k], Bs[col][k], row stride 34 floats (8B-aligned,
// conflict-free banks) so each fragment is a single aligned v2f load.
// A frag: lane L -> m=L&15, kh=(L>>4)*2, holds K=kh,kh+1.
// B frag: lane L -> n=L&15, holds K=kh,kh+1.
// D tile: lane L -> n=L&15, reg r -> m = r + 8*(L>>4).
// ------------------------------------------------------------------
#define KCH 32
#define LDP 34   // padded LDS row stride (floats)

template <int LAYER>
__global__ __launch_bounds__(256) void gemm_kernel(
    const float* __restrict__ Asrc,    // L0: data [ROWS,253]; L1: Y0 [ROWS,256]
    const float* __restrict__ Aaux0,   // L0: interp [ROWS,3]; L1: scale0 [256]
    const float* __restrict__ Aaux1,   // L0: unused;          L1: shift0 [256]
    const float* __restrict__ W,       // [outC, 256] row-major
    const float* __restrict__ bias,    // [outC]
    float* __restrict__ Y,             // [ROWS, outC]
    float* __restrict__ gsum,          // [outC]
    float* __restrict__ gsq,           // [outC]
    const int outC)
{
    __shared__ __align__(16) float As[128][LDP];  // rows x k
    __shared__ __align__(16) float Bs[128][LDP];  // cols x k (transposed)
    __shared__ float colsum[128];
    __shared__ float colsq[128];

    const int tid  = threadIdx.x;
    const int lane = tid & 31;
    const int wave = tid >> 5;
    const int wn   = wave & 1;        // N half
    const int wm   = wave >> 1;       // M quarter
    const int rowBase = blockIdx.x * 128;
    const int colBase = blockIdx.y * 128;

    if (tid < 128) { colsum[tid] = 0.0f; colsq[tid] = 0.0f; }

    v8f acc[2][4];
    #pragma unroll
    for (int f = 0; f < 2; ++f)
        #pragma unroll
        for (int t = 0; t < 4; ++t)
            #pragma unroll
            for (int r = 0; r < 8; ++r) acc[f][t][r] = 0.0f;

    const int m   = lane & 15;
    const int kh  = (lane >> 4) << 1;
    const int nl  = lane & 15;
    const int row0 = wm * 32 + m;
    const int row1 = row0 + 16;

    for (int kc = 0; kc < IN_CH; kc += KCH) {
        __syncthreads();
        // ---- stage A: 128 rows x 32 k (16 elems/thread, coalesced in k) ----
        #pragma unroll
        for (int i = 0; i < 16; ++i) {
            const int flat = tid + i * 256;      // 0..4095
            const int r  = flat >> 5;
            const int kk = flat & 31;
            const int c  = kc + kk;
            const size_t row = (size_t)(rowBase + r);
            float v;
            if (LAYER == 0) {
                v = (c < DATA_DIM) ? Asrc[row * DATA_DIM + c]
                                   : Aaux0[row * 3 + (c - DATA_DIM)];
            } else {
                const float y = Asrc[row * IN_CH + c];
                v = fmaxf(Aaux0[c] * y + Aaux1[c], 0.0f);
            }
            As[r][kk] = v;
        }
        // ---- stage B (transposed): Bs[nn][kk] = W[(colBase+nn)*256 + kc+kk] ----
        #pragma unroll
        for (int i = 0; i < 16; ++i) {
            const int flat = tid + i * 256;      // 0..4095
            const int nn = flat >> 5;
            const int kk = flat & 31;
            Bs[nn][kk] = W[(size_t)(colBase + nn) * IN_CH + kc + kk];
        }
        __syncthreads();
        // ---- WMMA over the k-chunk ----
        #pragma unroll
        for (int kk = 0; kk < KCH; kk += 4) {
            const v2f a0 = *reinterpret_cast<const v2f*>(&As[row0][kk + kh]);
            const v2f a1 = *reinterpret_cast<const v2f*>(&As[row1][kk + kh]);
            #pragma unroll
            for (int t = 0; t < 4; ++t) {
                const v2f b = *reinterpret_cast<const v2f*>(
                    &Bs[wn * 64 + t * 16 + nl][kk + kh]);
                acc[0][t] = __builtin_amdgcn_wmma_f32_16x16x4_f32(
                    false, a0, false, b, (short)0, acc[0][t], false, false);
                acc[1][t] = __builtin_amdgcn_wmma_f32_16x16x4_f32(
                    false, a1, false, b, (short)0, acc[1][t], false, false);
            }
        }
    }

    // ---- bias, store, per-column partial stats ----
    const int mh = (lane >> 4) << 3;  // +8 row offset for upper lane half
    #pragma unroll
    for (int t = 0; t < 4; ++t) {
        const int cLocal = wn * 64 + t * 16 + nl;
        const int c = colBase + cLocal;
        const float bv = bias[c];
        float s = 0.0f, sq = 0.0f;
        #pragma unroll
        for (int f = 0; f < 2; ++f) {
            #pragma unroll
            for (int r = 0; r < 8; ++r) {
                const float v = acc[f][t][r] + bv;
                const size_t row = (size_t)(rowBase + wm * 32 + f * 16 + r + mh);
                Y[row * (size_t)outC + c] = v;
                s  += v;
                sq += v * v;
            }
        }
        atomicAdd(&colsum[cLocal], s);
        atomicAdd(&colsq[cLocal], sq);
    }
    __syncthreads();
    if (tid < 128) {
        atomicAdd(&gsum[colBase + tid], colsum[tid]);
        atomicAdd(&gsq[colBase + tid],  colsq[tid]);
    }
}

// ------------------------------------------------------------------
// finalize BN: scale = g * rsqrt(var+eps); shift = be - mean*scale
// ------------------------------------------------------------------
__global__ void finalize_stats_kernel(
    const float* __restrict__ gsum, const float* __restrict__ gsq,
    const float* __restrict__ g, const float* __restrict__ be,
    float* __restrict__ scale, float* __restrict__ shift, int C)
{
    const int c = blockIdx.x * blockDim.x + threadIdx.x;
    if (c < C) {
        const float invN = 1.0f / (float)ROWS;
        const float mean = gsum[c] * invN;
        const float var  = gsq[c] * invN - mean * mean;
        const float sc   = g[c] * rsqrtf(var + BN_EPS);
        scale[c] = sc;
        shift[c] = be[c] - mean * sc;
    }
}

// ------------------------------------------------------------------
// final BN+ReLU applied in place on d_out (raw y1)
// ------------------------------------------------------------------
__global__ void bn_relu_kernel(float* __restrict__ out,
                               const float* __restrict__ scale,
                               const float* __restrict__ shift)
{
    const size_t i = (size_t)blockIdx.x * blockDim.x + threadIdx.x;
    const int c = (int)(i & (C1 - 1));
    const float v = out[i];
    out[i] = fmaxf(scale[c] * v + shift[c], 0.0f);
}

// ------------------------------------------------------------------
extern "C" void kernel_launch(void* const* d_in, const int* in_sizes, int n_in,
                              void* d_out, int out_size, void* d_ws, size_t ws_size,
                              hipStream_t stream) {
    const float* coords        = (const float*)d_in[0];
    const float* sample_coords = (const float*)d_in[1];
    const float* data          = (const float*)d_in[2];
    // d_in[3] = sample_data (unused by the reference)
    const float* W0  = (const float*)d_in[4];
    const float* b0  = (const float*)d_in[5];
    const float* g0  = (const float*)d_in[6];
    const float* be0 = (const float*)d_in[7];
    const float* W1  = (const float*)d_in[8];
    const float* b1  = (const float*)d_in[9];
    const float* g1  = (const float*)d_in[10];
    const float* be1 = (const float*)d_in[11];

    float* ws      = (float*)d_ws;
    float* interp  = ws + F_INTERP;
    float* sum0    = ws + F_SUM0;
    float* sq0     = ws + F_SQ0;
    float* scale0  = ws + F_SCALE0;
    float* shift0  = ws + F_SHIFT0;
    float* sum1    = ws + F_SUM1;
    float* sq1     = ws + F_SQ1;
    float* scale1  = ws + F_SCALE1;
    float* shift1  = ws + F_SHIFT1;
    float* Y0      = ws + F_Y0;
    float* out     = (float*)d_out;

    zero_stats_kernel<<<1, 256, 0, stream>>>(ws);

    interp_kernel<<<dim3(BATCH, NPTS / 128), 128, 0, stream>>>(
        coords, sample_coords, interp);

    gemm_kernel<0><<<dim3(ROWS / 128, C0 / 128), 256, 0, stream>>>(
        data, interp, nullptr, W0, b0, Y0, sum0, sq0, C0);

    finalize_stats_kernel<<<1, 256, 0, stream>>>(sum0, sq0, g0, be0,
                                                 scale0, shift0, C0);

    gemm_kernel<1><<<dim3(ROWS / 128, C1 / 128), 256, 0, stream>>>(
        Y0, scale0, shift0, W1, b1, out, sum1, sq1, C1);

    finalize_stats_kernel<<<1, 128, 0, stream>>>(sum1, sq1, g1, be1,
                                                 scale1, shift1, C1);

    bn_relu_kernel<<<(ROWS * C1) / 256, 256, 0, stream>>>(out, scale1, shift1);
}